// ExactTopKAttention_31928786878801
// MI455X (gfx1250) — compile-verified
//
#include <hip/hip_runtime.h>
#include <math.h>

typedef __attribute__((ext_vector_type(2))) float v2f;
typedef __attribute__((ext_vector_type(4))) float v4f;
typedef __attribute__((ext_vector_type(8))) float v8f;

#define N_BATCH 2
#define LSEQ    2048
#define HD      512
#define NH      8
#define ED      64
#define TOPK    32
#define TEMP    0.125f   // 1/sqrt(64)

// ---------------------------------------------------------------------------
// Projection GEMM: out = in @ W.T + b, 16x16 tiles via V_WMMA_F32_16X16X4_F32.
// in: [4096, 512] row-major. W: [512, 512] row-major (W[o][i]).
// mode 0/2: write head-major [N,H,L,E]; mode 1: write transposed [N,H,E,L].
// These outputs ARE re-read (hot) -> default RT cache policy keeps them in L2.
// ---------------------------------------------------------------------------
__global__ __launch_bounds__(256) void proj_kernel(
    const float* __restrict__ in, const float* __restrict__ W,
    const float* __restrict__ bias, float* __restrict__ out, int mode)
{
    const int waveId = threadIdx.x >> 5;
    const int lane   = threadIdx.x & 31;
    const int tile   = blockIdx.x * 8 + waveId;   // 8192 tiles total
    const int ct     = tile & 31;                 // col tile (512/16)
    const int rt     = tile >> 5;                 // row tile (4096/16)

    const int m  = lane & 15;   // A row / D column lane index
    const int kh = lane >> 4;   // K half select (per 16x4 f32 A layout)

    const int rowBase = rt * 16;
    const int col     = ct * 16 + m;              // output feature o

    const float* arow = in + (size_t)(rowBase + m) * HD;
    const float* brow = W  + (size_t)col * HD;    // B[k][n] = W[o=col][i=k]

    v8f acc = {};
#pragma unroll 8
    for (int kb = 0; kb < HD; kb += 4) {
        const int k = kb + 2 * kh;
        v2f a = *(const v2f*)(arow + k);
        v2f b = *(const v2f*)(brow + k);
        acc = __builtin_amdgcn_wmma_f32_16x16x4_f32(
            false, a, false, b, (short)0, acc, false, false);
    }

    const float bcol = bias[col];
    const int h = col >> 6, e = col & 63;
#pragma unroll
    for (int v = 0; v < 8; ++v) {
        const int gr = rowBase + v + 8 * kh;      // global row in [0,4096)
        const int nn = gr >> 11;                  // batch
        const int l  = gr & (LSEQ - 1);
        const float d = acc[v] + bcol;
        if (mode == 1)
            out[(((size_t)nn * NH + h) * ED + e) * LSEQ + l] = d;   // kT
        else
            out[(((size_t)nn * NH + h) * LSEQ + l) * ED + e] = d;   // qh / vh
    }
}

// ---------------------------------------------------------------------------
// Fused top-k attention. One block = one (n, h, 16-row stripe of L).
// Phase 1: WMMA QK^T into a 16x2048 LDS stripe (128 KB, possible only with
//          CDNA5's 320 KB/WGP LDS -> QK never round-trips HBM).
// Phase 2: exact top-32 per row (wave argmax-remove, one pick per lane),
//          softmax, dense-A write (NT stores: 268 MB streaming, keep L2 for
//          the reused K/V working set), sparse AV accumulate.
// ---------------------------------------------------------------------------
__global__ __launch_bounds__(256) void attn_kernel(
    const float* __restrict__ qh, const float* __restrict__ kT,
    const float* __restrict__ vh, float* __restrict__ Vout,
    float* __restrict__ Aout)
{
    extern __shared__ float lds[];                // [16][2048] scores

    const int waveId = threadIdx.x >> 5;
    const int lane   = threadIdx.x & 31;
    const int nh     = blockIdx.x >> 7;           // n*8 + h
    const int rt     = blockIdx.x & 127;          // row tile within L
    const int n      = nh >> 3, h = nh & 7;

    const int m  = lane & 15;
    const int kh = lane >> 4;

    // ---- Phase 1: scores = temp * (q_tile @ kT) -------------------------
    const float* qbase = qh + ((size_t)nh * LSEQ + rt * 16) * ED;
    v2f afrag[16];                                // shared A tile, hoisted
#pragma unroll
    for (int s = 0; s < 16; ++s) {
        const int e = 4 * s + 2 * kh;
        afrag[s] = *(const v2f*)(qbase + (size_t)m * ED + e);
    }

    const float* kbase = kT + (size_t)nh * ED * LSEQ;
    for (int ct = waveId; ct < 128; ct += 8) {
        const int col = ct * 16 + m;
        // prefetch next column tile of kT into cache
        if (ct + 8 < 128)
            __builtin_prefetch(kbase + col + 128, 0, 1);
        v8f acc = {};
#pragma unroll
        for (int s = 0; s < 16; ++s) {
            const int e = 4 * s + 2 * kh;
            v2f b;
            b.x = kbase[(size_t)e * LSEQ + col];
            b.y = kbase[(size_t)(e + 1) * LSEQ + col];
            acc = __builtin_amdgcn_wmma_f32_16x16x4_f32(
                false, afrag[s], false, b, (short)0, acc, false, false);
        }
#pragma unroll
        for (int v = 0; v < 8; ++v)
            lds[(v + 8 * kh) * LSEQ + col] = acc[v] * TEMP;
    }
    __syncthreads();

    // ---- Phase 2: top-32 + softmax + outputs (2 rows per wave) ----------
    float* Abase = Aout + ((size_t)nh * LSEQ + rt * 16) * LSEQ;
    const float* vb = vh + (size_t)nh * LSEQ * ED;

    for (int t = 0; t < 2; ++t) {
        const int r = waveId * 2 + t;
        const float* srow = lds + (size_t)r * LSEQ;

        float vals[64];
#pragma unroll
        for (int i = 0; i < 64; ++i) vals[i] = srow[lane + 32 * i];

        float lmax = -INFINITY; int lidx = 0;
#pragma unroll
        for (int i = 0; i < 64; ++i)
            if (vals[i] > lmax) { lmax = vals[i]; lidx = i; }

        float selv = -INFINITY; int selc = 0;   // lane j holds round-j pick
        for (int j = 0; j < TOPK; ++j) {
            float gmax = lmax;
#pragma unroll
            for (int off = 16; off; off >>= 1)
                gmax = fmaxf(gmax, __shfl_xor(gmax, off, 32));
            const unsigned long long bal = __ballot(lmax == gmax);
            const int w = (int)__ffsll(bal) - 1;
            const int wcol = __shfl(lidx, w, 32) * 32 + w;
            if (lane == j) { selv = gmax; selc = wcol; }
            if (lane == w) {
#pragma unroll
                for (int i = 0; i < 64; ++i)
                    if (i == lidx) vals[i] = -INFINITY;
                lmax = -INFINITY; lidx = 0;
#pragma unroll
                for (int i = 0; i < 64; ++i)
                    if (vals[i] > lmax) { lmax = vals[i]; lidx = i; }
            }
        }

        // softmax over the 32 selected values (one per lane)
        const float rmax = __shfl(selv, 0, 32);
        const float ev = __expf(selv - rmax);
        float Z = ev;
#pragma unroll
        for (int off = 16; off; off >>= 1) Z += __shfl_xor(Z, off, 32);
        const float p = ev / Z;

        // dense A row: NT zero-fill (vectorized) then NT scatter of the 32
        // probabilities (same-wave stores stay ordered; NT only changes the
        // cache temporal policy so this 268 MB stream doesn't sweep L2).
        float* Arow = Abase + (size_t)r * LSEQ;
        const v4f z4 = {0.f, 0.f, 0.f, 0.f};
#pragma unroll
        for (int i = 0; i < 16; ++i)
            __builtin_nontemporal_store(z4, (v4f*)Arow + i * 32 + lane);
        __builtin_nontemporal_store(p, Arow + selc);

        // sparse AV: each lane owns dims (lane, lane+32) of E=64
        float acc0 = 0.f, acc1 = 0.f;
#pragma unroll
        for (int j = 0; j < TOPK; ++j) {
            const float pj = __shfl(p, j, 32);
            const int   cj = __shfl(selc, j, 32);
            const float* vrow = vb + (size_t)cj * ED;
            acc0 += pj * vrow[lane];
            acc1 += pj * vrow[lane + 32];
        }
        const int l = rt * 16 + r;
        float* orow = Vout + ((size_t)n * LSEQ + l) * HD + h * ED;
        __builtin_nontemporal_store(acc0, orow + lane);
        __builtin_nontemporal_store(acc1, orow + lane + 32);
    }
}

// ---------------------------------------------------------------------------
extern "C" void kernel_launch(void* const* d_in, const int* in_sizes, int n_in,
                              void* d_out, int out_size, void* d_ws, size_t ws_size,
                              hipStream_t stream) {
    const float* queries = (const float*)d_in[0];
    const float* keys    = (const float*)d_in[1];
    const float* values  = (const float*)d_in[2];
    const float* Wq = (const float*)d_in[3];
    const float* bq = (const float*)d_in[4];
    const float* Wk = (const float*)d_in[5];
    const float* bk = (const float*)d_in[6];
    const float* Wv = (const float*)d_in[7];
    const float* bv = (const float*)d_in[8];

    float* out = (float*)d_out;
    float* Vout = out;                                   // [2,2048,512]
    float* Aout = out + (size_t)N_BATCH * LSEQ * HD;     // [2,8,2048,2048]

    const size_t projElems = (size_t)N_BATCH * LSEQ * HD;  // 2,097,152
    float* qh = (float*)d_ws;            // [N,H,L,E]
    float* kT = qh + projElems;          // [N,H,E,L]
    float* vh = kT + projElems;          // [N,H,L,E]

    dim3 blk(256);
    // 8192 tiles (256 row-tiles x 32 col-tiles), 8 waves/block
    proj_kernel<<<1024, blk, 0, stream>>>(queries, Wq, bq, qh, 0);
    proj_kernel<<<1024, blk, 0, stream>>>(keys,    Wk, bk, kT, 1);
    proj_kernel<<<1024, blk, 0, stream>>>(values,  Wv, bv, vh, 2);

    // one block per (n, h, 16-row stripe): 2*8*128 = 2048 blocks, 128 KB LDS
    attn_kernel<<<2048, blk, 16 * LSEQ * sizeof(float), stream>>>(
        qh, kT, vh, Vout, Aout);
}